// MDGTDTInet_28853590295304
// MI455X (gfx1250) — compile-verified
//
#include <hip/hip_runtime.h>
#include <math.h>

// MDGTDTInet forward for MI455X (gfx1250, wave32, WMMA fp32 16x16x4).
// B=8, LP=1000, NC=64, D=128, E=320.

typedef __attribute__((ext_vector_type(2))) float v2f;
typedef __attribute__((ext_vector_type(8))) float v8f;

#define WAVES_PER_BLOCK 8

// ---------------------------------------------------------------------------
// Generic batched GEMM: C[b] = act( alpha * A[b] @ opB(B[b]) + bias )
//   A: M x K row-major (optionally ReLU'd on load)
//   B: K x N row-major (TRANS_B=0)  or  N x K row-major as B^T (TRANS_B=1)
// One wave computes a 16(M) x 64(N) strip = 4 WMMA accumulators, sharing the
// A fragment across all 4 (4x arithmetic intensity vs one tile/wave).
// Edge handling is branch-free: out-of-range rows/cols are clamped to a valid
// address and the loaded fragment is multiplied by a 0/1 mask, so the K loop
// is straight-line and every WMMA executes with full EXEC per the ISA.
// Addresses are carried as INTEGER offsets from the kernel-arg base pointers
// (not loop-carried pointers) so clang's address-space inference keeps all
// loads on the global path (global_load_*), avoiding flat_load_* which would
// tick DScnt and force s_wait_loadcnt_dscnt before each WMMA.
// Fragment layouts per CDNA5 ISA 7.12.2:
//   A 16x4 : lanes 0-15 -> M=lane, VGPR{0,1}=K{0,1}; lanes 16-31 -> K{2,3}
//   B 4x16 : lanes 0-15 -> N=lane, VGPR{0,1}=K{0,1}; lanes 16-31 -> K{2,3}
//   C 16x16: VGPR r -> M=r (lanes 0-15), M=r+8 (lanes 16-31), N=lane&15
// ---------------------------------------------------------------------------
template <bool TRANS_B, bool RELU_A, bool RELU_OUT, bool HAS_BIAS>
__global__ __launch_bounds__(256) void gemm_wmma_kernel(
    const float* __restrict__ A, const float* __restrict__ Bm,
    const float* __restrict__ bias, float* __restrict__ C,
    int M, int N, int K,
    long long sA, long long sB, long long sC,
    float alpha, int stripsN, long long tilesPerBatch, long long totalWaves) {
  long long gwave = (long long)blockIdx.x * WAVES_PER_BLOCK + (threadIdx.x >> 5);
  if (gwave >= totalWaves) return;

  const int lane = threadIdx.x & 31;
  const int half = lane >> 4;   // 0: K pair {0,1}, 1: K pair {2,3}
  const int idx  = lane & 15;   // M index for A frag, N index for B frag

  long long bidx = gwave / tilesPerBatch;
  long long t    = gwave - bidx * tilesPerBatch;
  int tm = (int)(t / stripsN);
  int tn = (int)(t - (long long)tm * stripsN);

  const long long baseA = bidx * sA;
  const long long baseB = bidx * sB;
  const long long baseC = bidx * sC;

  const int row0 = tm << 4;  // 16 rows
  const int col0 = tn << 6;  // 64 cols (4 x 16)

  // ---- A side: clamped row + mask (branch-free edge handling) ----
  const int mrow = row0 + idx;
  const float amask = (mrow < M) ? 1.0f : 0.0f;
  const int mclamp = (mrow < M) ? mrow : (M - 1);
  long long aoff = baseA + (long long)mclamp * K + half * 2;

  // ---- B side: 4 sub-tiles, each with clamped col + mask ----
  long long boff[4];
  float nmask[4];
#pragma unroll
  for (int j = 0; j < 4; ++j) {
    const int n = col0 + j * 16 + idx;
    nmask[j] = (n < N) ? 1.0f : 0.0f;
    const int nc = (n < N) ? n : (N - 1);
    boff[j] = TRANS_B ? (baseB + (long long)nc * K + half * 2)
                      : (baseB + (long long)(half * 2) * N + nc);
  }

  v8f acc0 = {}, acc1 = {}, acc2 = {}, acc3 = {};
#pragma unroll 2
  for (int k = 0; k < K; k += 4) {  // K is always a multiple of 4 here
    v2f av = *(const v2f*)(A + aoff);
    aoff += 4;
    av.x *= amask;
    av.y *= amask;
    if (RELU_A) {
      av.x = fmaxf(av.x, 0.f);
      av.y = fmaxf(av.y, 0.f);
    }
    v2f bv[4];
#pragma unroll
    for (int j = 0; j < 4; ++j) {
      if (TRANS_B) {
        bv[j] = *(const v2f*)(Bm + boff[j]);
        boff[j] += 4;
      } else {
        bv[j].x = Bm[boff[j]];
        bv[j].y = Bm[boff[j] + N];
        boff[j] += (long long)4 * N;
      }
      bv[j].x *= nmask[j];
      bv[j].y *= nmask[j];
    }
    acc0 = __builtin_amdgcn_wmma_f32_16x16x4_f32(false, av, false, bv[0],
                                                 (short)0, acc0, false, false);
    acc1 = __builtin_amdgcn_wmma_f32_16x16x4_f32(false, av, false, bv[1],
                                                 (short)0, acc1, false, false);
    acc2 = __builtin_amdgcn_wmma_f32_16x16x4_f32(false, av, false, bv[2],
                                                 (short)0, acc2, false, false);
    acc3 = __builtin_amdgcn_wmma_f32_16x16x4_f32(false, av, false, bv[3],
                                                 (short)0, acc3, false, false);
  }

  v8f accs[4] = {acc0, acc1, acc2, acc3};
#pragma unroll
  for (int j = 0; j < 4; ++j) {
    const int n = col0 + j * 16 + idx;
    if (n < N) {
      const float bval = HAS_BIAS ? bias[n] : 0.f;
#pragma unroll
      for (int r = 0; r < 8; ++r) {
        const int mm = row0 + half * 8 + r;
        if (mm < M) {
          float v = accs[j][r] * alpha + bval;
          if (RELU_OUT) v = fmaxf(v, 0.f);
          C[baseC + (long long)mm * N + n] = v;
        }
      }
    }
  }
}

template <bool TB, bool RA, bool RO, bool BI>
static void launch_gemm(hipStream_t s, const float* A, const float* Bm,
                        const float* bias, float* C, int M, int N, int K,
                        int batch, long long sA, long long sB, long long sC,
                        float alpha) {
  const int tilesM = (M + 15) >> 4;
  const int stripsN = (N + 63) >> 6;
  const long long tpb = (long long)tilesM * stripsN;
  const long long total = (long long)batch * tpb;
  const int blocks = (int)((total + WAVES_PER_BLOCK - 1) / WAVES_PER_BLOCK);
  gemm_wmma_kernel<TB, RA, RO, BI><<<blocks, 256, 0, s>>>(
      A, Bm, bias, C, M, N, K, sA, sB, sC, alpha, stripsN, tpb, total);
}

// ---------------------------------------------------------------------------
// Row softmax over ncols (one 256-thread block per row).
// ---------------------------------------------------------------------------
__global__ __launch_bounds__(256) void softmax_rows_kernel(float* __restrict__ S,
                                                           int ncols) {
  float* p = S + (long long)blockIdx.x * ncols;
  const int tid = threadIdx.x;
  __shared__ float red[8];

  float m = -3.402823466e38f;
  for (int j = tid; j < ncols; j += 256) m = fmaxf(m, p[j]);
  for (int off = 16; off; off >>= 1) m = fmaxf(m, __shfl_xor(m, off, 32));
  if ((tid & 31) == 0) red[tid >> 5] = m;
  __syncthreads();
  float bm = fmaxf(fmaxf(fmaxf(red[0], red[1]), fmaxf(red[2], red[3])),
                   fmaxf(fmaxf(red[4], red[5]), fmaxf(red[6], red[7])));
  __syncthreads();

  float s = 0.f;
  for (int j = tid; j < ncols; j += 256) {
    float e = __expf(p[j] - bm);
    p[j] = e;
    s += e;
  }
  for (int off = 16; off; off >>= 1) s += __shfl_xor(s, off, 32);
  if ((tid & 31) == 0) red[tid >> 5] = s;
  __syncthreads();
  float bs = red[0] + red[1] + red[2] + red[3] + red[4] + red[5] + red[6] + red[7];
  const float inv = 1.0f / bs;
  for (int j = tid; j < ncols; j += 256) p[j] *= inv;
}

// ---------------------------------------------------------------------------
// In-place sigmoid over [B, perBatch]; per-batch sum accumulated atomically.
// grid = (chunks, B), block = 256.
// ---------------------------------------------------------------------------
__global__ __launch_bounds__(256) void sigmoid_sum_kernel(float* __restrict__ A,
                                                          float* __restrict__ sums,
                                                          int perBatch) {
  const int b = blockIdx.y;
  float* p = A + (long long)b * perBatch;
  float local = 0.f;
  for (int j = blockIdx.x * blockDim.x + threadIdx.x; j < perBatch;
       j += gridDim.x * blockDim.x) {
    float v = 1.0f / (1.0f + __expf(-p[j]));
    p[j] = v;
    local += v;
  }
  for (int off = 16; off; off >>= 1) local += __shfl_xor(local, off, 32);
  __shared__ float red[8];
  if ((threadIdx.x & 31) == 0) red[threadIdx.x >> 5] = local;
  __syncthreads();
  if (threadIdx.x < 8) {
    float v = red[threadIdx.x];
    for (int off = 4; off; off >>= 1) v += __shfl_xor(v, off, 32);
    if (threadIdx.x == 0) atomicAdd(&sums[b], v);
  }
}

// ---------------------------------------------------------------------------
// cp[b][d] += sum_{i in block's chunk, k} tanh(pf[b,i,d]*cf[b,k,d]) * A[b,i,k]
// Compound feats (64x128 = 32KB) cached in LDS; A row staged per i.
// grid = (LP/iPerBlock, B), block = 128 (one thread per d).
// ---------------------------------------------------------------------------
__global__ __launch_bounds__(128) void cp_embed_kernel(
    const float* __restrict__ pf, const float* __restrict__ cf,
    const float* __restrict__ Asig, float* __restrict__ cp, int LP,
    int iPerBlock) {
  const int NC = 64, D = 128;
  __shared__ float cf_s[NC * D];
  __shared__ float a_s[NC];
  const int b = blockIdx.y;
  const int d = threadIdx.x;

  const float* cfb = cf + (long long)b * NC * D;
  for (int j = d; j < NC * D; j += 128) cf_s[j] = cfb[j];
  __syncthreads();

  const float* pfb = pf + (long long)b * LP * D;
  const float* Ab = Asig + (long long)b * LP * NC;
  const int i0 = blockIdx.x * iPerBlock;
  const int i1 = min(i0 + iPerBlock, LP);

  float acc = 0.f;
  for (int i = i0; i < i1; ++i) {
    __syncthreads();
    if (d < NC) a_s[d] = Ab[(long long)i * NC + d];
    __syncthreads();
    const float p = pfb[(long long)i * D + d];
#pragma unroll 8
    for (int k = 0; k < NC; ++k) acc += tanhf(p * cf_s[k * D + d]) * a_s[k];
  }
  atomicAdd(&cp[b * D + d], acc);
}

// cp[b][d] /= sums[b]  (folds the A-normalization into cp_embedding)
__global__ __launch_bounds__(128) void scale_cp_kernel(float* __restrict__ cp,
                                                       const float* __restrict__ sums) {
  cp[blockIdx.x * 128 + threadIdx.x] *= (1.0f / sums[blockIdx.x]);
}

// out[b] = sum_d h3[b,d]*w[d] + bias   (N=1 final layer)
__global__ __launch_bounds__(256) void final_dot_kernel(
    const float* __restrict__ h3, const float* __restrict__ w,
    const float* __restrict__ bptr, float* __restrict__ out, int K) {
  const int b = blockIdx.x, tid = threadIdx.x;
  float v = (tid < K) ? h3[(long long)b * K + tid] * w[tid] : 0.f;
  for (int off = 16; off; off >>= 1) v += __shfl_xor(v, off, 32);
  __shared__ float red[8];
  if ((tid & 31) == 0) red[tid >> 5] = v;
  __syncthreads();
  if (tid == 0) {
    float s = 0.f;
    for (int i = 0; i < 8; ++i) s += red[i];
    out[b] = s + bptr[0];
  }
}

// ---------------------------------------------------------------------------
extern "C" void kernel_launch(void* const* d_in, const int* in_sizes, int n_in,
                              void* d_out, int out_size, void* d_ws,
                              size_t ws_size, hipStream_t stream) {
  (void)in_sizes; (void)n_in; (void)out_size; (void)ws_size;
  const int B = 8, LP = 1000, NC = 64, D = 128, E = 320;

  const float* pemb = (const float*)d_in[0];   // [B,LP,E]
  const float* pfx  = (const float*)d_in[1];   // [B,LP,D]
  const float* cf   = (const float*)d_in[2];   // [B,NC,D]
  const float* emb_w = (const float*)d_in[3];  const float* emb_b = (const float*)d_in[4];
  const float* wq_w  = (const float*)d_in[5];  const float* wq_b  = (const float*)d_in[6];
  const float* wk_w  = (const float*)d_in[7];  const float* wk_b  = (const float*)d_in[8];
  const float* wv_w  = (const float*)d_in[9];  const float* wv_b  = (const float*)d_in[10];
  const float* wo_w  = (const float*)d_in[11]; const float* wo_b  = (const float*)d_in[12];
  const float* jp_w  = (const float*)d_in[13]; const float* jp_b  = (const float*)d_in[14];
  const float* jc_w  = (const float*)d_in[15]; const float* jc_b  = (const float*)d_in[16];
  const float* c1_w  = (const float*)d_in[17]; const float* c1_b  = (const float*)d_in[18];
  const float* c2_w  = (const float*)d_in[19]; const float* c2_b  = (const float*)d_in[20];
  const float* c3_w  = (const float*)d_in[21]; const float* c3_b  = (const float*)d_in[22];
  const float* c4_w  = (const float*)d_in[23]; const float* c4_b  = (const float*)d_in[24];
  float* out = (float*)d_out;

  char* ws = (char*)d_ws;
  size_t off = 0;
  auto alloc = [&](size_t nfloats) -> float* {
    size_t o = off;
    off += (nfloats * sizeof(float) + 255) & ~(size_t)255;
    return (float*)(ws + o);
  };
  float* pe   = alloc((size_t)B * LP * D);   // embedded protein
  float* Qb   = alloc((size_t)B * LP * D);
  float* Kb   = alloc((size_t)B * LP * D);
  float* Vb   = alloc((size_t)B * LP * D);
  float* S    = alloc((size_t)B * LP * LP);  // attention scores / probs (32 MB)
  float* O    = alloc((size_t)B * LP * D);   // attn @ V
  float* pf   = alloc((size_t)B * LP * D);   // protein_feats
  float* P    = alloc((size_t)B * LP * D);
  float* Cc   = alloc((size_t)B * NC * D);
  float* Alog = alloc((size_t)B * LP * NC);  // sigmoid joint attention
  float* cp   = alloc((size_t)B * D);
  float* sums = alloc((size_t)B);
  float* h1   = alloc((size_t)B * 1024);
  float* h2   = alloc((size_t)B * 1024);
  float* h3   = alloc((size_t)B * 256);

  // deterministic zero-init of accumulators every call
  hipMemsetAsync(cp, 0, (size_t)B * D * sizeof(float), stream);
  hipMemsetAsync(sums, 0, (size_t)B * sizeof(float), stream);

  const long long sPD = (long long)LP * D;   // per-batch stride for [LP,D]
  const long long sSS = (long long)LP * LP;  // per-batch stride for [LP,LP]
  const long long sCD = (long long)NC * D;   // per-batch stride for [NC,D]
  const long long sPN = (long long)LP * NC;  // per-batch stride for [LP,NC]
  const float invSqrtD = 0.08838834764831845f;  // 1/sqrt(128)

  // 1) pe = protein_embedding @ emb_w + emb_b   (M=8000, K=320, N=128)
  launch_gemm<false, false, false, true>(stream, pemb, emb_w, emb_b, pe,
                                         B * LP, D, E, 1, 0, 0, 0, 1.0f);
  // 2) Q/K/V projections (M=8000, K=128, N=128)
  launch_gemm<false, false, false, true>(stream, pe, wq_w, wq_b, Qb,
                                         B * LP, D, D, 1, 0, 0, 0, 1.0f);
  launch_gemm<false, false, false, true>(stream, pfx, wk_w, wk_b, Kb,
                                         B * LP, D, D, 1, 0, 0, 0, 1.0f);
  launch_gemm<false, false, false, true>(stream, pfx, wv_w, wv_b, Vb,
                                         B * LP, D, D, 1, 0, 0, 0, 1.0f);
  // 3) S = (Q @ K^T) / sqrt(D)   batched NT, M=N=1000, K=128
  launch_gemm<true, false, false, false>(stream, Qb, Kb, nullptr, S,
                                         LP, LP, D, B, sPD, sPD, sSS, invSqrtD);
  // 4) row softmax over 1000
  softmax_rows_kernel<<<B * LP, 256, 0, stream>>>(S, LP);
  // 5) O = attn @ V   batched NN, M=1000, N=128, K=1000
  launch_gemm<false, false, false, false>(stream, S, Vb, nullptr, O,
                                          LP, D, LP, B, sSS, sPD, sPD, 1.0f);
  // 6) protein_feats = O @ wo + b
  launch_gemm<false, false, false, true>(stream, O, wo_w, wo_b, pf,
                                         B * LP, D, D, 1, 0, 0, 0, 1.0f);
  // 7) P = relu(pf) @ jp + b ;  Cc = relu(cf) @ jc + b
  launch_gemm<false, true, false, true>(stream, pf, jp_w, jp_b, P,
                                        B * LP, D, D, 1, 0, 0, 0, 1.0f);
  launch_gemm<false, true, false, true>(stream, cf, jc_w, jc_b, Cc,
                                        B * NC, D, D, 1, 0, 0, 0, 1.0f);
  // 8) joint attention logits = P @ Cc^T   batched NT, M=1000, N=64, K=128
  launch_gemm<true, false, false, false>(stream, P, Cc, nullptr, Alog,
                                         LP, NC, D, B, sPD, sCD, sPN, 1.0f);
  // 9) A = sigmoid(logits); sums[b] = sum(A[b])
  sigmoid_sum_kernel<<<dim3(64, B), 256, 0, stream>>>(Alog, sums, LP * NC);
  // 10) cp[b,d] = sum_{i,k} tanh(pf*cf) * A ; then divide by sums[b]
  const int iPerBlock = 50;
  cp_embed_kernel<<<dim3((LP + iPerBlock - 1) / iPerBlock, B), 128, 0, stream>>>(
      pf, cf, Alog, cp, LP, iPerBlock);
  scale_cp_kernel<<<B, 128, 0, stream>>>(cp, sums);
  // 11) classifier MLP
  launch_gemm<false, false, true, true>(stream, cp, c1_w, c1_b, h1,
                                        B, 1024, D, 1, 0, 0, 0, 1.0f);
  launch_gemm<false, false, true, true>(stream, h1, c2_w, c2_b, h2,
                                        B, 1024, 1024, 1, 0, 0, 0, 1.0f);
  launch_gemm<false, false, true, true>(stream, h2, c3_w, c3_b, h3,
                                        B, 256, 1024, 1, 0, 0, 0, 1.0f);
  final_dot_kernel<<<B, 256, 0, stream>>>(h3, c4_w, c4_b, out, 256);
}